// SubspaceRankHopLayer_10849087390027
// MI455X (gfx1250) — compile-verified
//
#include <hip/hip_runtime.h>

#define N_NODES 8192
#define IN_DIM  256
#define OUT_DIM 256
#define KHOPS   5
#define SDIM    64
#define ETA     0.5f
#define LAM_LAP 0.3f
#define LAM_SP  0.05f
#define EPS_C   0.5f
#define LN_EPS  1e-5f

typedef __attribute__((ext_vector_type(2))) float v2f;
typedef __attribute__((ext_vector_type(8))) float v8f;

__device__ __forceinline__ unsigned lds_addr32(const void* p) {
    // generic shared pointer: low 32 bits are the wave-relative LDS byte offset
    return (unsigned)(unsigned long long)p;
}

// ---------------------------------------------------------------------------
// 1) CholeskyQR per hop: A = U^T U, A = L L^T, Q = U L^-T   (one block per hop)
// ---------------------------------------------------------------------------
__global__ void cholqr_kernel(const float* __restrict__ U, float* __restrict__ Qw) {
    __shared__ float A[SDIM * SDIM];
    const int k = blockIdx.x;
    const float* Uk = U + (size_t)k * IN_DIM * SDIM;
    float* Qk = Qw + (size_t)k * IN_DIM * SDIM;
    const int tid = threadIdx.x; // 256 threads

    for (int e = tid; e < SDIM * SDIM; e += 256) {
        int i = e / SDIM, j = e % SDIM;
        float acc = 0.f;
        for (int d = 0; d < IN_DIM; ++d) acc += Uk[d * SDIM + i] * Uk[d * SDIM + j];
        A[e] = acc;
    }
    __syncthreads();

    for (int j = 0; j < SDIM; ++j) {
        if (tid == 0) {
            float v = A[j * SDIM + j];
            for (int s = 0; s < j; ++s) v -= A[j * SDIM + s] * A[j * SDIM + s];
            A[j * SDIM + j] = sqrtf(fmaxf(v, 1e-20f));
        }
        __syncthreads();
        if (tid > j && tid < SDIM) {
            float v = A[tid * SDIM + j];
            for (int s = 0; s < j; ++s) v -= A[tid * SDIM + s] * A[j * SDIM + s];
            A[tid * SDIM + j] = v / A[j * SDIM + j];
        }
        __syncthreads();
    }

    {
        const int r = tid;
        float q[SDIM];
        for (int j = 0; j < SDIM; ++j) {
            float v = Uk[r * SDIM + j];
            for (int s = 0; s < j; ++s) v -= q[s] * A[j * SDIM + s];
            v /= A[j * SDIM + j];
            q[j] = v;
            Qk[r * SDIM + j] = v;
        }
    }
}

// ---------------------------------------------------------------------------
// 2) P[k][n][s] = sum_d hops[k][n][d] * Q[k][d][s]
// ---------------------------------------------------------------------------
__global__ void proj_kernel(const float* __restrict__ hops, const float* __restrict__ Qw,
                            float* __restrict__ P) {
    size_t idx = (size_t)blockIdx.x * blockDim.x + threadIdx.x;
    const int s = (int)(idx % SDIM);
    size_t nk = idx / SDIM;
    const int n = (int)(nk % N_NODES);
    const int k = (int)(nk / N_NODES);
    const float* h = hops + ((size_t)k * N_NODES + n) * IN_DIM;
    const float* q = Qw + (size_t)k * IN_DIM * SDIM + s;
    float acc = 0.f;
    for (int d = 0; d < IN_DIM; ++d) acc += h[d] * q[d * SDIM];
    P[idx] = acc;
}

// ---------------------------------------------------------------------------
// 3) G[k][i][j] = sum_n P[k][n][i] * P[k][n][j]
// ---------------------------------------------------------------------------
__global__ void gram_kernel(const float* __restrict__ P, float* __restrict__ G) {
    int idx = blockIdx.x * blockDim.x + threadIdx.x;
    const int j = idx % SDIM;
    const int i = (idx / SDIM) % SDIM;
    const int k = idx / (SDIM * SDIM);
    const float* Pk = P + (size_t)k * N_NODES * SDIM;
    float acc = 0.f;
    for (int n = 0; n < N_NODES; ++n) acc += Pk[(size_t)n * SDIM + i] * Pk[(size_t)n * SDIM + j];
    G[idx] = acc;
}

// ---------------------------------------------------------------------------
// 4) rank_eff per hop, M = I + c*G Cholesky, softmax weights. Writes out tail.
// ---------------------------------------------------------------------------
__global__ void stats_kernel(const float* __restrict__ G, float* __restrict__ Mchol,
                             float* __restrict__ wbuf, float* __restrict__ out_tail,
                             const float* __restrict__ log_tau) {
    __shared__ float re[KHOPS];
    const int k = threadIdx.x;
    if (k < KHOPS) {
        const float* Gk = G + k * SDIM * SDIM;
        const float inv_n = 1.0f / ((float)N_NODES + 1e-8f);
        float tr = 0.f, tr2 = 0.f;
        for (int i = 0; i < SDIM; ++i) tr += Gk[i * SDIM + i];
        tr *= inv_n;
        for (int e = 0; e < SDIM * SDIM; ++e) { float v = Gk[e] * inv_n; tr2 += v * v; }
        re[k] = tr * tr / (tr2 + 1e-8f);

        const float coeff = (float)SDIM / ((float)N_NODES * EPS_C * EPS_C + 1e-8f);
        float* Mk = Mchol + k * SDIM * SDIM;
        for (int e = 0; e < SDIM * SDIM; ++e) {
            int i = e / SDIM, j = e % SDIM;
            Mk[e] = coeff * Gk[e] + (i == j ? 1.f : 0.f);
        }
        for (int j = 0; j < SDIM; ++j) {
            float v = Mk[j * SDIM + j];
            for (int s = 0; s < j; ++s) v -= Mk[j * SDIM + s] * Mk[j * SDIM + s];
            const float d = sqrtf(fmaxf(v, 1e-20f));
            Mk[j * SDIM + j] = d;
            for (int i = j + 1; i < SDIM; ++i) {
                float u = Mk[i * SDIM + j];
                for (int s = 0; s < j; ++s) u -= Mk[i * SDIM + s] * Mk[j * SDIM + s];
                Mk[i * SDIM + j] = u / d;
            }
        }
    }
    __syncthreads();
    if (k == 0) {
        float tau = __expf(log_tau[0]);
        tau = fminf(fmaxf(tau, 0.1f), 10.f);
        float mx = -1e30f;
        for (int i = 0; i < KHOPS; ++i) mx = fmaxf(mx, re[i] / tau);
        float wv[KHOPS], ssum = 0.f;
        for (int i = 0; i < KHOPS; ++i) { wv[i] = __expf(re[i] / tau - mx); ssum += wv[i]; }
        for (int i = 0; i < KHOPS; ++i) {
            const float wi = wv[i] / ssum;
            wbuf[i] = wi;
            out_tail[i] = re[i];
            out_tail[KHOPS + i] = wi;
        }
    }
}

// ---------------------------------------------------------------------------
// 5) P row <- w_k * M_k^-1 * P row   (cho_solve per row, L in LDS)
// ---------------------------------------------------------------------------
__global__ void solve_kernel(float* __restrict__ P, const float* __restrict__ Mchol,
                             const float* __restrict__ wbuf) {
    __shared__ float Lm[SDIM * SDIM];
    const int k = blockIdx.y;
    for (int e = threadIdx.x; e < SDIM * SDIM; e += blockDim.x) Lm[e] = Mchol[k * SDIM * SDIM + e];
    __syncthreads();
    const float wk = wbuf[k];
    const int n = blockIdx.x * blockDim.x + threadIdx.x;
    float* p = P + ((size_t)k * N_NODES + n) * SDIM;
    float y[SDIM];
    for (int i = 0; i < SDIM; ++i) {
        float v = p[i];
        for (int s = 0; s < i; ++s) v -= Lm[i * SDIM + s] * y[s];
        y[i] = v / Lm[i * SDIM + i];
    }
    for (int i = SDIM - 1; i >= 0; --i) {
        float v = y[i];
        for (int s = i + 1; s < SDIM; ++s) v -= Lm[s * SDIM + i] * y[s];
        y[i] = v / Lm[i * SDIM + i];
    }
    for (int i = 0; i < SDIM; ++i) p[i] = wk * y[i];
}

// ---------------------------------------------------------------------------
// 6) grad_sum[n][d] = sum_k sum_s P[k][n][s] * Q[k][d][s]
// ---------------------------------------------------------------------------
__global__ void gradsum_kernel(const float* __restrict__ P, const float* __restrict__ Qw,
                               float* __restrict__ Gsum) {
    size_t idx = (size_t)blockIdx.x * blockDim.x + threadIdx.x;
    const int d = (int)(idx % IN_DIM);
    const size_t n = idx / IN_DIM;
    float acc = 0.f;
    for (int k = 0; k < KHOPS; ++k) {
        const float* p = P + ((size_t)k * N_NODES + n) * SDIM;
        const float* q = Qw + ((size_t)k * IN_DIM + d) * SDIM;
        for (int s = 0; s < SDIM; ++s) acc += p[s] * q[s];
    }
    Gsum[idx] = acc;
}

// ---------------------------------------------------------------------------
// 7) Hbar = sum_k w_k * hops_k
// ---------------------------------------------------------------------------
__global__ void hbar_kernel(const float* __restrict__ hops, const float* __restrict__ wbuf,
                            float* __restrict__ Hbar) {
    size_t idx = (size_t)blockIdx.x * blockDim.x + threadIdx.x;
    float acc = 0.f;
    for (int k = 0; k < KHOPS; ++k)
        acc += wbuf[k] * hops[(size_t)k * N_NODES * IN_DIM + idx];
    Hbar[idx] = acc;
}

// ---------------------------------------------------------------------------
// 8) lap = L @ Hbar via V_WMMA_F32_16X16X4_F32 with async-to-LDS staging.
//    One block = 16 rows x full 256 cols  =>  L streamed from HBM exactly once.
//    8 waves; wave w owns col tiles w*16 and w*16+128 (two accumulators).
//    A chunk (16 x KC) and B chunk (KC x 256) staged in LDS via
//    GLOBAL_LOAD_ASYNC_TO_LDS (ASYNCcnt), drained with s_wait_asynccnt.
// ---------------------------------------------------------------------------
#define KC    32
#define ASTR  34    // padded A row stride (floats): conflict-free, 8B aligned
#define BSTR  260   // padded B row stride (floats): conflict-free, 16B aligned

__global__ void lap_wmma_kernel(const float* __restrict__ Lm, const float* __restrict__ Hbar,
                                const float* __restrict__ H, const float* __restrict__ Gsum,
                                float* __restrict__ Hprox) {
    __shared__ float sA[16 * ASTR];
    __shared__ float sB[KC * BSTR];

    const int tid  = threadIdx.x;
    const int lane = tid & 31;
    const int wave = tid >> 5;
    const int row0 = blockIdx.x * 16;
    const int m    = lane & 15;          // M index (A) / N index (B)
    const int koff = (lane >> 4) << 1;   // 0 or 2 per ISA 16x4 f32 fragment layout

    // --- async staging thread mapping -------------------------------------
    // A chunk: 16 rows x KC floats = 256 float2 -> one b64 per thread
    const int arow = tid >> 4;           // 0..15
    const int ac2  = (tid & 15) << 1;    // float column (even)
    unsigned long long gA = (unsigned long long)(Lm + (size_t)(row0 + arow) * N_NODES + ac2);
    const unsigned lA = lds_addr32(&sA[arow * ASTR + ac2]);

    // B chunk: KC rows x 256 floats = 2048 float4 -> eight b128 per thread
    unsigned long long gB[8];
    unsigned lB[8];
#pragma unroll
    for (int i = 0; i < 8; ++i) {
        const int fidx = tid + i * 256;
        const int brow = fidx >> 6;          // 0..31
        const int bc4  = fidx & 63;          // float4 column
        gB[i] = (unsigned long long)(Hbar + (size_t)brow * IN_DIM + bc4 * 4);
        lB[i] = lds_addr32(&sB[brow * BSTR + bc4 * 4]);
    }

    v8f c0 = {}, c1 = {};
    for (int k = 0; k < N_NODES; k += KC) {
        asm volatile("global_load_async_to_lds_b64 %0, %1, off"
                     :: "v"(lA), "v"(gA) : "memory");
#pragma unroll
        for (int i = 0; i < 8; ++i) {
            asm volatile("global_load_async_to_lds_b128 %0, %1, off"
                         :: "v"(lB[i]), "v"(gB[i]) : "memory");
        }
        gA += (unsigned long long)KC * 4ull;
#pragma unroll
        for (int i = 0; i < 8; ++i) gB[i] += (unsigned long long)KC * IN_DIM * 4ull;

        asm volatile("s_wait_asynccnt 0x0" ::: "memory");
        __syncthreads();

#pragma unroll
        for (int kk = 0; kk < KC; kk += 4) {
            v2f a;
            a.x = sA[m * ASTR + kk + koff];
            a.y = sA[m * ASTR + kk + koff + 1];
            v2f b0, b1;
            b0.x = sB[(kk + koff) * BSTR + wave * 16 + m];
            b0.y = sB[(kk + koff + 1) * BSTR + wave * 16 + m];
            b1.x = sB[(kk + koff) * BSTR + 128 + wave * 16 + m];
            b1.y = sB[(kk + koff + 1) * BSTR + 128 + wave * 16 + m];
            c0 = __builtin_amdgcn_wmma_f32_16x16x4_f32(false, a, false, b0, (short)0, c0, false, false);
            c1 = __builtin_amdgcn_wmma_f32_16x16x4_f32(false, a, false, b1, (short)0, c1, false, false);
        }
        __syncthreads();
    }

    // epilogue: H_half = H + ETA*Gsum - ETA*LAM_LAP*lap ; soft-threshold
    const int nn = lane & 15;
    const int mbase = (lane >> 4) * 8;
    const int col0a = wave * 16;
    const int col0b = wave * 16 + 128;
#pragma unroll
    for (int g = 0; g < 8; ++g) {
        const int r = row0 + mbase + g;
        {
            const size_t off = (size_t)r * IN_DIM + (col0a + nn);
            const float h = H[off] + ETA * Gsum[off] - ETA * LAM_LAP * c0[g];
            const float s = fabsf(h) - LAM_SP * ETA;
            Hprox[off] = s > 0.f ? copysignf(s, h) : 0.f;
        }
        {
            const size_t off = (size_t)r * IN_DIM + (col0b + nn);
            const float h = H[off] + ETA * Gsum[off] - ETA * LAM_LAP * c1[g];
            const float s = fabsf(h) - LAM_SP * ETA;
            Hprox[off] = s > 0.f ? copysignf(s, h) : 0.f;
        }
    }
}

// ---------------------------------------------------------------------------
// 9) Lin = Hprox @ W_in^T via V_WMMA_F32_16X16X4_F32
// ---------------------------------------------------------------------------
__global__ void out_wmma_kernel(const float* __restrict__ Hprox, const float* __restrict__ Win,
                                float* __restrict__ Lin) {
    const int lane = threadIdx.x & 31;
    const int wave = threadIdx.x >> 5;
    const int row0 = (blockIdx.x * 8 + wave) * 16;
    const int col0 = blockIdx.y * 16;
    const int m    = lane & 15;
    const int koff = (lane >> 4) << 1;

    v8f c = {};
    const float* arow = Hprox + (size_t)(row0 + m) * IN_DIM + koff;
    const float* brow = Win + (size_t)(col0 + m) * IN_DIM + koff; // B[kk][o]=W_in[o][kk]
    for (int kk = 0; kk < IN_DIM; kk += 4) {
        v2f a = *(const v2f*)(arow + kk);
        v2f b = *(const v2f*)(brow + kk);
        c = __builtin_amdgcn_wmma_f32_16x16x4_f32(false, a, false, b, (short)0, c, false, false);
    }

    const int nn = lane & 15;
    const int mbase = (lane >> 4) * 8;
#pragma unroll
    for (int g = 0; g < 8; ++g) {
        const int r = row0 + mbase + g;
        Lin[(size_t)r * OUT_DIM + (col0 + nn)] = c[g];
    }
}

// ---------------------------------------------------------------------------
// 10) LayerNorm per row -> d_out
// ---------------------------------------------------------------------------
__global__ void ln_kernel(const float* __restrict__ Lin, const float* __restrict__ gamma,
                          const float* __restrict__ beta, float* __restrict__ out) {
    __shared__ float red[256];
    const int r = blockIdx.x;
    const int t = threadIdx.x;
    const float v = Lin[(size_t)r * OUT_DIM + t];
    red[t] = v;
    __syncthreads();
    for (int s = 128; s > 0; s >>= 1) { if (t < s) red[t] += red[t + s]; __syncthreads(); }
    const float mu = red[0] / (float)OUT_DIM;
    __syncthreads();
    const float dv = v - mu;
    red[t] = dv * dv;
    __syncthreads();
    for (int s = 128; s > 0; s >>= 1) { if (t < s) red[t] += red[t + s]; __syncthreads(); }
    const float var = red[0] / (float)OUT_DIM;
    out[(size_t)r * OUT_DIM + t] = dv * rsqrtf(var + LN_EPS) * gamma[t] + beta[t];
}

// ---------------------------------------------------------------------------
extern "C" void kernel_launch(void* const* d_in, const int* in_sizes, int n_in,
                              void* d_out, int out_size, void* d_ws, size_t ws_size,
                              hipStream_t stream) {
    const float* H    = (const float*)d_in[0];
    const float* hops = (const float*)d_in[1];
    const float* L    = (const float*)d_in[2];
    const float* U    = (const float*)d_in[3];
    const float* Win  = (const float*)d_in[4];
    const float* gam  = (const float*)d_in[5];
    const float* bet  = (const float*)d_in[6];
    const float* ltau = (const float*)d_in[7];
    float* out = (float*)d_out;
    float* ws  = (float*)d_ws;

    // workspace layout (floats)
    float* Qw    = ws;                                       // K*IN_DIM*SDIM   = 81920
    float* P     = Qw + (size_t)KHOPS * IN_DIM * SDIM;       // K*N*SDIM        = 2621440
    float* G     = P + (size_t)KHOPS * N_NODES * SDIM;       // K*SDIM*SDIM     = 20480
    float* Mchol = G + (size_t)KHOPS * SDIM * SDIM;          //                 = 20480
    float* wbuf  = Mchol + (size_t)KHOPS * SDIM * SDIM;      //                 = 16 (pad)
    float* Hbar  = wbuf + 16;                                // N*IN_DIM        = 2097152 (reused as Lin)
    float* Gsum  = Hbar + (size_t)N_NODES * IN_DIM;          //                 = 2097152
    float* Hprox = Gsum + (size_t)N_NODES * IN_DIM;          //                 = 2097152

    float* out_tail = out + (size_t)N_NODES * OUT_DIM;

    cholqr_kernel<<<KHOPS, 256, 0, stream>>>(U, Qw);

    {   // P = hops @ Q
        size_t total = (size_t)KHOPS * N_NODES * SDIM;
        proj_kernel<<<(unsigned)(total / 256), 256, 0, stream>>>(hops, Qw, P);
    }
    gram_kernel<<<(KHOPS * SDIM * SDIM) / 256, 256, 0, stream>>>(P, G);
    stats_kernel<<<1, 64, 0, stream>>>(G, Mchol, wbuf, out_tail, ltau);
    {   // P <- w_k * M^-1 P
        dim3 grid(N_NODES / 256, KHOPS);
        solve_kernel<<<grid, 256, 0, stream>>>(P, Mchol, wbuf);
    }
    gradsum_kernel<<<(N_NODES * IN_DIM) / 256, 256, 0, stream>>>(P, Qw, Gsum);
    hbar_kernel<<<(N_NODES * IN_DIM) / 256, 256, 0, stream>>>(hops, wbuf, Hbar);

    // lap = L @ Hbar (WMMA f32, async-LDS staged) + fused prox epilogue;
    // one block per 16-row strip covering all 256 cols -> L read once.
    lap_wmma_kernel<<<N_NODES / 16, 256, 0, stream>>>(L, Hbar, H, Gsum, Hprox);

    {   // Lin = Hprox @ W_in^T (WMMA f32); reuse Hbar buffer as Lin
        dim3 grid(N_NODES / 128, OUT_DIM / 16);
        out_wmma_kernel<<<grid, 256, 0, stream>>>(Hprox, Win, Hbar);
    }
    ln_kernel<<<N_NODES, 256, 0, stream>>>(Hbar, gam, bet, out);
}